// Gridsampler_57690000720485
// MI455X (gfx1250) — compile-verified
//
#include <hip/hip_runtime.h>
#include <hip/hip_bf16.h>

// ---------------------------------------------------------------------------
// Deformable grid-sampler (B=4, C=3, H=W=512, DS=4 -> h=w=128, K=13 -> 169 taps)
// Memory-bound gather kernel. CDNA5 path: async global->LDS staging of the
// three streamed arrays (kernels / offsets_h / offsets_v), depth-4 pipeline,
// tracked with ASYNCcnt (s_wait_asynccnt). Uniform (per-block) image/output
// bases so gathers lower to saddr-form global_load with 32-bit voffsets and
// constant plane offsets folded into the 24-bit instruction offset.
// ---------------------------------------------------------------------------

namespace {
constexpr int KS   = 13;          // kernel side
constexpr int K2   = KS * KS;     // 169 taps
constexpr int PADV = KS / 2;      // 6
constexpr int IH   = 512;
constexpr int IW   = 512;
constexpr int IHW  = IH * IW;
constexpr int OH   = 128;
constexpr int OW   = 128;
constexpr int OHW  = OH * OW;
constexpr int NC   = 3;
constexpr int PF   = 4;           // pipeline depth (power of 2)
constexpr int TPB  = 128;         // threads per block (4 waves on wave32)
constexpr int HPAD = IH + 2 * PADV;  // 524
constexpr int WPAD = IW + 2 * PADV;  // 524
constexpr int BLOCKS_PER_B = OHW / TPB;  // 128
}

// ---- gfx1250 async global->LDS copy + ASYNCcnt wait ------------------------
typedef __attribute__((address_space(1))) int gint_as1_t;
typedef __attribute__((address_space(3))) int lint_as3_t;

__device__ __forceinline__ void async_copy_f32(const float* g, float* l) {
#if __has_builtin(__builtin_amdgcn_global_load_async_to_lds_b32)
  __builtin_amdgcn_global_load_async_to_lds_b32(
      (gint_as1_t*)(void*)g, (lint_as3_t*)(void*)l, /*offset=*/0, /*cpol=*/0);
#else
  unsigned lofs = (unsigned)(__SIZE_TYPE__)l;
  asm volatile("global_load_async_to_lds_b32 %0, %1, off"
               :
               : "v"(lofs), "v"(g)
               : "memory");
#endif
}

#if __has_builtin(__builtin_amdgcn_s_wait_asynccnt)
#define WAIT_ASYNC(n) __builtin_amdgcn_s_wait_asynccnt(n)
#else
#define WAIT_ASYNC(n) asm volatile("s_wait_asynccnt %0" ::"n"(n) : "memory")
#endif

// reflect a padded-image coordinate back into the original image (single
// reflection is sufficient: padded idx in [0, 523], PAD=6, N=512)
__device__ __forceinline__ int reflect_idx(int i) {
  int j = i - PADV;
  j = (j < 0) ? -j : j;
  j = (j >= IW) ? (2 * IW - 2 - j) : j;
  return j;
}

__global__ __launch_bounds__(TPB) void Gridsampler_57690000720485_kernel(
    const float* __restrict__ img,     // (B, 3, 512, 512)
    const float* __restrict__ kern,    // (B, 169, 128, 128)
    const float* __restrict__ offh,    // (B, 169, 128, 128)
    const float* __restrict__ offv,    // (B, 169, 128, 128)
    const float* __restrict__ unitp,   // scalar
    float* __restrict__ out)           // (B, 3, 128, 128)
{
  // staged {kernel, off_h, off_v, pad} per thread; 16B/thread -> ds_load_b128
  __shared__ __align__(16) float stg[PF][TPB][4];

  const int tid = threadIdx.x;
  const int b   = blockIdx.x >> 7;                      // uniform: 128 blk / batch
  const int rem = ((blockIdx.x & (BLOCKS_PER_B - 1)) << 7) | tid;  // 0..16383
  const int y   = rem >> 7;
  const int x   = rem & (OW - 1);

  const float unit = unitp[0];

  // streamed-array pointers for this pixel's tap 0 (stride OHW between taps)
  const long base_in = (long)(b * K2) * OHW + rem;
  const float* kp = kern + base_in;
  const float* hp = offh + base_in;
  const float* vp = offv + base_in;

  // prime the pipeline: stages 0..PF-2  (3*(PF-1) = 9 outstanding async ops)
#pragma unroll
  for (int s = 0; s < PF - 1; ++s) {
    const long o = (long)s * OHW;
    async_copy_f32(kp + o, &stg[s][tid][0]);
    async_copy_f32(hp + o, &stg[s][tid][1]);
    async_copy_f32(vp + o, &stg[s][tid][2]);
  }

  const float bx = (float)(x * 4);
  const float by = (float)(y * 4);
  const float* imgb = img + (long)(b * NC) * IHW;       // uniform -> SGPR base

  float acc0 = 0.f, acc1 = 0.f, acc2 = 0.f;

  int   kx  = 0;
  float kxf = 0.f, kyf = 0.f;

  auto compute_tap = [&](int buf) {
    const float4 t = *(const float4*)(&stg[buf][tid][0]);  // one ds_load_b128
    const float kw = t.x;
    const float oh = t.y;
    const float ov = t.z;

    const float px = bx + kxf + oh * unit;
    const float py = by + kyf + ov * unit;
    const float fx = floorf(px);
    const float fy = floorf(py);
    const float a  = px - fx;
    const float bt = py - fy;

    int xL = (int)fx; xL = xL < 0 ? 0 : (xL > WPAD - 1 ? WPAD - 1 : xL);
    int xR = xL + 1;  xR = xR > WPAD - 1 ? WPAD - 1 : xR;
    int yT = (int)fy; yT = yT < 0 ? 0 : (yT > HPAD - 1 ? HPAD - 1 : yT);
    int yB = yT + 1;  yB = yB > HPAD - 1 ? HPAD - 1 : yB;

    // padded coords -> original image coords (reflect padding, never stored)
    const int cL = reflect_idx(xL);
    const int cR = reflect_idx(xR);
    const int rT = reflect_idx(yT) * IW;
    const int rB = reflect_idx(yB) * IW;

    const int iTL = rT + cL;
    const int iTR = rT + cR;
    const int iBL = rB + cL;
    const int iBR = rB + cR;

    // bilinear weights folded with the kernel weight (shared by all channels)
    const float w11  = a * bt;
    const float w01  = (a - w11) * kw;
    const float w10  = (bt - w11) * kw;
    const float w00  = (1.f - a - bt + w11) * kw;
    const float w11k = w11 * kw;

    // plane offsets (IHW, 2*IHW elements) are constants -> fold into IOFFSET
    acc0 += w00 * imgb[iTL]           + w01 * imgb[iTR] +
            w10 * imgb[iBL]           + w11k * imgb[iBR];
    acc1 += w00 * imgb[iTL + IHW]     + w01 * imgb[iTR + IHW] +
            w10 * imgb[iBL + IHW]     + w11k * imgb[iBR + IHW];
    acc2 += w00 * imgb[iTL + 2 * IHW] + w01 * imgb[iTR + 2 * IHW] +
            w10 * imgb[iBL + 2 * IHW] + w11k * imgb[iBR + 2 * IHW];

    ++kx; kxf += 1.f;
    if (kx == KS) { kx = 0; kxf = 0.f; kyf += 1.f; }
  };

  // main loop: issue stage k+PF-1, wait for stage k, compute tap k
  for (int k = 0; k <= K2 - PF; ++k) {
    const int nb = (k + PF - 1) & (PF - 1);
    const long o = (long)(k + PF - 1) * OHW;
    async_copy_f32(kp + o, &stg[nb][tid][0]);
    async_copy_f32(hp + o, &stg[nb][tid][1]);
    async_copy_f32(vp + o, &stg[nb][tid][2]);
    WAIT_ASYNC(3 * (PF - 1));       // oldest stage (k) has landed
    asm volatile("" ::: "memory");  // keep LDS reads after the wait
    compute_tap(k & (PF - 1));
  }

  // drain: last PF-1 taps are already in flight / landed
  WAIT_ASYNC(0);
  asm volatile("" ::: "memory");
#pragma unroll
  for (int k = K2 - PF + 1; k < K2; ++k) {
    compute_tap(k & (PF - 1));
  }

  float* outb = out + (long)(b * NC) * OHW;             // uniform -> SGPR base
  outb[rem]           = acc0;
  outb[rem + OHW]     = acc1;
  outb[rem + 2 * OHW] = acc2;
}

extern "C" void kernel_launch(void* const* d_in, const int* in_sizes, int n_in,
                              void* d_out, int out_size, void* d_ws, size_t ws_size,
                              hipStream_t stream) {
  (void)in_sizes; (void)n_in; (void)out_size; (void)d_ws; (void)ws_size;
  const float* img  = (const float*)d_in[0];
  const float* kern = (const float*)d_in[1];
  const float* oh   = (const float*)d_in[2];
  const float* ov   = (const float*)d_in[3];
  const float* unit = (const float*)d_in[4];
  float*       out  = (float*)d_out;

  const int total  = 4 * OHW;          // B * h * w = 65536 pixels
  const int blocks = total / TPB;      // 512
  hipLaunchKernelGGL(Gridsampler_57690000720485_kernel,
                     dim3(blocks), dim3(TPB), 0, stream,
                     img, kern, oh, ov, unit, out);
}